// GraphCN_66271345377540
// MI455X (gfx1250) — compile-verified
//
#include <hip/hip_runtime.h>

typedef float v2f __attribute__((ext_vector_type(2)));
typedef float v8f __attribute__((ext_vector_type(8)));

__global__ void k_zero_f32(float* __restrict__ p, unsigned n) {
  unsigned i = blockIdx.x * blockDim.x + threadIdx.x;
  if (i < n) p[i] = 0.0f;
}

__global__ void k_degrees(const int* __restrict__ src, const int* __restrict__ dst,
                          float* __restrict__ outdeg, float* __restrict__ indeg, unsigned E) {
  unsigned e = blockIdx.x * blockDim.x + threadIdx.x;
  if (e < E) {
    atomicAdd(&outdeg[src[e]], 1.0f);
    atomicAdd(&indeg[dst[e]], 1.0f);
  }
}

__global__ void k_inv_sqrt_clip(float* __restrict__ p, unsigned n) {
  unsigned i = blockIdx.x * blockDim.x + threadIdx.x;
  if (i < n) {
    float d = p[i];
    d = d < 1.0f ? 1.0f : d;
    p[i] = rsqrtf(d);
  }
}

// agg[dst[e]][0:F] += feat[src[e]][0:F] * oscale[src[e]], F = 1<<shift.
// Each thread handles one float4 chunk: coalesced b128 gather, 4 f32 atomics
// (one atomic per aggregated element is irreducible). Buffers are L2-resident.
__global__ void k_edge_scatter4(const float* __restrict__ feat, const float* __restrict__ oscale,
                                const int* __restrict__ src, const int* __restrict__ dst,
                                float* __restrict__ agg, unsigned E, int shift) {
  const int s2 = shift - 2;                     // float4 chunks per row
  unsigned idx = blockIdx.x * blockDim.x + threadIdx.x;
  unsigned q = idx & ((1u << s2) - 1u);
  unsigned e = idx >> s2;
  if (e < E) {
    int s = src[e];
    int d = dst[e];
    float sc = oscale[s];
    const float4 v = ((const float4*)(feat + ((size_t)s << shift)))[q];
    float* ap = agg + ((size_t)d << shift) + ((size_t)q << 2);
    atomicAdd(ap + 0, v.x * sc);
    atomicAdd(ap + 1, v.y * sc);
    atomicAdd(ap + 2, v.z * sc);
    atomicAdd(ap + 3, v.w * sc);
  }
}

// C(MxN) = A(MxK) @ B(KxN) via V_WMMA_F32_16X16X4_F32, K,N compile-time.
// Block = 8 waves; whole B staged in LDS (transposed, stride K+4 => ds_load_b64
// pairs, conflict-free banks). Each wave owns a 16-row x N strip: one A-frag
// global load feeds NT=N/16 back-to-back WMMAs on independent accumulators.
// Epilogue fuses per-row scale (D_in^-1/2), bias, optional ReLU.
template <int K, int N>
__global__ void k_gemm_wmma_f32(const float* __restrict__ A, const float* __restrict__ B,
                                const float* __restrict__ rowscale, const float* __restrict__ bias,
                                float* __restrict__ C, int M, int relu) {
  constexpr int NT  = N / 16;
  constexpr int LDB = K + 4;                    // pad: bank-conflict-free ds reads
  __shared__ float sB[N * LDB];
  for (int i = threadIdx.x; i < K * N; i += blockDim.x) {
    int kk = i / N, nn = i % N;                 // N is pow2 -> shifts
    sB[nn * LDB + kk] = B[i];
  }
  __syncthreads();

  const int lane = threadIdx.x & 31;
  const int wave = threadIdx.x >> 5;
  const int m0 = (blockIdx.x * (blockDim.x >> 5) + wave) << 4;
  if (m0 >= M) return;                          // wave-uniform: EXEC all-ones
  const int l16 = lane & 15;
  const int hi  = lane >> 4;                    // lanes 0-15 vs 16-31
  const int kb  = hi << 1;                      // frag VGPR0 K=kb, VGPR1 K=kb+1
  const float* Ap = A + (size_t)(m0 + l16) * K + kb;
  const float* Bp = &sB[l16 * LDB + kb];

  v8f c[NT];
#pragma unroll
  for (int t = 0; t < NT; ++t) c[t] = {};

#pragma unroll 4
  for (int k = 0; k < K; k += 4) {
    v2f a = *(const v2f*)(Ap + k);              // global_load_b64, imm offset
#pragma unroll
    for (int t = 0; t < NT; ++t) {
      v2f b = *(const v2f*)(Bp + t * 16 * LDB + k);   // ds_load_b64, imm offset
      c[t] = __builtin_amdgcn_wmma_f32_16x16x4_f32(false, a, false, b, (short)0,
                                                   c[t], false, false);
    }
  }

  const int mb = hi << 3;                       // C/D: VGPR j holds row mb+j
  float bv[NT];
#pragma unroll
  for (int t = 0; t < NT; ++t) bv[t] = bias ? bias[l16 + t * 16] : 0.0f;
#pragma unroll
  for (int j = 0; j < 8; ++j) {
    const int row = m0 + mb + j;
    const float rs = rowscale ? rowscale[row] : 1.0f;
    float* Crow = C + (size_t)row * N + l16;
#pragma unroll
    for (int t = 0; t < NT; ++t) {
      float v = c[t][j] * rs + bv[t];
      if (relu) v = fmaxf(v, 0.0f);
      Crow[t * 16] = v;
    }
  }
}

// out[i] = out[i] * iscale[row] + bias[f]
__global__ void k_scale_bias(float* __restrict__ out, const float* __restrict__ iscale,
                             const float* __restrict__ bias, unsigned total, int shift) {
  unsigned i = blockIdx.x * blockDim.x + threadIdx.x;
  if (i < total) {
    unsigned row = i >> shift;
    unsigned f = i & ((1u << shift) - 1u);
    out[i] = out[i] * iscale[row] + bias[f];
  }
}

static inline size_t align256(size_t x) { return (x + 255) & ~(size_t)255; }

extern "C" void kernel_launch(void* const* d_in, const int* in_sizes, int n_in,
                              void* d_out, int out_size, void* d_ws, size_t ws_size,
                              hipStream_t stream) {
  const float* x  = (const float*)d_in[0];
  const float* W1 = (const float*)d_in[1];
  const float* b1 = (const float*)d_in[2];
  const float* W2 = (const float*)d_in[3];
  const float* b2 = (const float*)d_in[4];
  const int*  src = (const int*)d_in[5];
  const int*  dst = (const int*)d_in[6];

  const int IN = 128, HF = 128, NC = 64;
  const int Nn = in_sizes[0] / IN;          // 50000
  const unsigned E = (unsigned)in_sizes[5]; // 800000
  float* out = (float*)d_out;               // doubles as layer-2 agg buffer

  char* ws = (char*)d_ws;
  float* oscale = (float*)ws; ws += align256((size_t)Nn * 4);
  float* iscale = (float*)ws; ws += align256((size_t)Nn * 4);
  float* agg1   = (float*)ws; ws += align256((size_t)Nn * IN * 4);
  float* h      = (float*)ws; ws += align256((size_t)Nn * HF * 4);
  float* y      = (float*)ws; ws += align256((size_t)Nn * NC * 4);
  (void)ws_size; (void)n_in; (void)out_size;

  const int T = 256;
  auto nblk = [](unsigned n, int t) { return (n + (unsigned)t - 1) / (unsigned)t; };

  // degrees -> D_out^-1/2 (oscale), D_in^-1/2 (iscale); shared by both layers
  k_zero_f32<<<nblk(Nn, T), T, 0, stream>>>(oscale, Nn);
  k_zero_f32<<<nblk(Nn, T), T, 0, stream>>>(iscale, Nn);
  k_degrees<<<nblk(E, T), T, 0, stream>>>(src, dst, oscale, iscale, E);
  k_inv_sqrt_clip<<<nblk(Nn, T), T, 0, stream>>>(oscale, Nn);
  k_inv_sqrt_clip<<<nblk(Nn, T), T, 0, stream>>>(iscale, Nn);

  // layer 1: agg1 = scatter(x * oscale); h = relu(iscale ⊙ (agg1 @ W1) + b1)
  k_zero_f32<<<nblk((unsigned)Nn * IN, T), T, 0, stream>>>(agg1, (unsigned)Nn * IN);
  k_edge_scatter4<<<nblk(E << 5, T), T, 0, stream>>>(x, oscale, src, dst, agg1, E, 7);
  {
    int mtiles = (Nn + 15) >> 4;            // 3125 waves of 16 rows
    k_gemm_wmma_f32<128, 128><<<(mtiles + 7) / 8, 256, 0, stream>>>(
        agg1, W1, iscale, b1, h, Nn, 1);
  }

  // layer 2: GEMM first (128 -> 64 feats) to halve edge traffic, then aggregate:
  //   y = h @ W2;  out = iscale ⊙ scatter(y * oscale) + b2
  {
    int mtiles = (Nn + 15) >> 4;
    k_gemm_wmma_f32<128, 64><<<(mtiles + 7) / 8, 256, 0, stream>>>(
        h, W2, nullptr, nullptr, y, Nn, 0);
  }
  k_zero_f32<<<nblk((unsigned)Nn * NC, T), T, 0, stream>>>(out, (unsigned)Nn * NC);
  k_edge_scatter4<<<nblk(E << 4, T), T, 0, stream>>>(y, oscale, src, dst, out, E, 6);
  k_scale_bias<<<nblk((unsigned)Nn * NC, T), T, 0, stream>>>(out, iscale, b2,
                                                             (unsigned)Nn * NC, 6);
}